// BalanceLabelAugmentation2_5265629905348
// MI455X (gfx1250) — compile-verified
//
#include <hip/hip_runtime.h>
#include <hip/hip_bf16.h>
#include <math.h>

typedef __attribute__((ext_vector_type(2))) float v2f;
typedef __attribute__((ext_vector_type(8))) float v8f;

#define N_O 16384
#define N_U 16384
#define DD  1024
#define NC  51
#define NCP 64

// ---------- kernel 1: build Wcat[128 x 1024] = [pad64(W); pad64(W_o)] ----------
__global__ __launch_bounds__(256) void k_pad_weights(const float* __restrict__ W,
                                                     const float* __restrict__ W_o,
                                                     float* __restrict__ Wcat) {
    int i = blockIdx.x * 256 + threadIdx.x;      // 0 .. 128*1024-1
    int row = i >> 10;
    int k   = i & (DD - 1);
    float v = 0.0f;
    if (row < NC)                    v = W[row * DD + k];
    else if (row >= 64 && row < 64 + NC) v = W_o[(row - 64) * DD + k];
    Wcat[i] = v;
}

// ---------- kernel 2: o-rows: G[64 cols] = A[M x 1024] @ Wcat[0:64]^T ----------
// block = 128 threads = 4 waves; wave w computes N-tile w; block covers 64 rows.
__global__ __launch_bounds__(128) void k_gemm_wmma(const float* __restrict__ A,
                                                   const float* __restrict__ Bp,
                                                   float* __restrict__ G) {
    const int lane  = threadIdx.x & 31;
    const int ntile = threadIdx.x >> 5;          // 0..3
    const int mbase = blockIdx.x * 64;
    const int rlo   = lane & 15;
    const int khalf = (lane >> 4) << 1;          // 0 or 2

    const float* a0 = A + (size_t)(mbase + rlo) * DD + khalf;
    const float* a1 = a0 + 16 * DD;
    const float* a2 = a0 + 32 * DD;
    const float* a3 = a0 + 48 * DD;
    const float* bp = Bp + (size_t)(ntile * 16 + rlo) * DD + khalf;

    v8f acc0 = {}; v8f acc1 = {}; v8f acc2 = {}; v8f acc3 = {};
#pragma unroll 8
    for (int k = 0; k < DD; k += 4) {
        v2f bv  = *(const v2f*)(bp + k);
        v2f av0 = *(const v2f*)(a0 + k);
        v2f av1 = *(const v2f*)(a1 + k);
        v2f av2 = *(const v2f*)(a2 + k);
        v2f av3 = *(const v2f*)(a3 + k);
        acc0 = __builtin_amdgcn_wmma_f32_16x16x4_f32(false, av0, false, bv, (short)0, acc0, false, false);
        acc1 = __builtin_amdgcn_wmma_f32_16x16x4_f32(false, av1, false, bv, (short)0, acc1, false, false);
        acc2 = __builtin_amdgcn_wmma_f32_16x16x4_f32(false, av2, false, bv, (short)0, acc2, false, false);
        acc3 = __builtin_amdgcn_wmma_f32_16x16x4_f32(false, av3, false, bv, (short)0, acc3, false, false);
    }

    const int col = ntile * 16 + rlo;
    const int rhi = (lane >> 4) << 3;            // 0 or 8
    float* g = G + (size_t)(mbase + rhi) * NCP + col;
#pragma unroll
    for (int r = 0; r < 8; ++r) {
        g[(size_t)(r)      * NCP] = acc0[r];
        g[(size_t)(r + 16) * NCP] = acc1[r];
        g[(size_t)(r + 32) * NCP] = acc2[r];
        g[(size_t)(r + 48) * NCP] = acc3[r];
    }
}

// ---------- kernel 3: u-rows: both heads in one pass over feat_u ----------
// block = 256 threads = 8 waves; waves 0-3 -> W logits (Gw), waves 4-7 -> W_o logits (Go).
__global__ __launch_bounds__(256) void k_gemm_wmma_dual(const float* __restrict__ A,
                                                        const float* __restrict__ Wcat,
                                                        float* __restrict__ Gw_u,
                                                        float* __restrict__ Go) {
    const int lane  = threadIdx.x & 31;
    const int ntile = threadIdx.x >> 5;          // 0..7 (cols ntile*16 .. +15 of Wcat)
    const int mbase = blockIdx.x * 64;
    const int rlo   = lane & 15;
    const int khalf = (lane >> 4) << 1;

    const float* a0 = A + (size_t)(mbase + rlo) * DD + khalf;
    const float* a1 = a0 + 16 * DD;
    const float* a2 = a0 + 32 * DD;
    const float* a3 = a0 + 48 * DD;
    const float* bp = Wcat + (size_t)(ntile * 16 + rlo) * DD + khalf;

    v8f acc0 = {}; v8f acc1 = {}; v8f acc2 = {}; v8f acc3 = {};
#pragma unroll 8
    for (int k = 0; k < DD; k += 4) {
        v2f bv  = *(const v2f*)(bp + k);
        v2f av0 = *(const v2f*)(a0 + k);
        v2f av1 = *(const v2f*)(a1 + k);
        v2f av2 = *(const v2f*)(a2 + k);
        v2f av3 = *(const v2f*)(a3 + k);
        acc0 = __builtin_amdgcn_wmma_f32_16x16x4_f32(false, av0, false, bv, (short)0, acc0, false, false);
        acc1 = __builtin_amdgcn_wmma_f32_16x16x4_f32(false, av1, false, bv, (short)0, acc1, false, false);
        acc2 = __builtin_amdgcn_wmma_f32_16x16x4_f32(false, av2, false, bv, (short)0, acc2, false, false);
        acc3 = __builtin_amdgcn_wmma_f32_16x16x4_f32(false, av3, false, bv, (short)0, acc3, false, false);
    }

    const int colg = ntile * 16 + rlo;           // 0..127 global col in Wcat
    const int rhi  = (lane >> 4) << 3;
    float* g = (ntile < 4)
                 ? (Gw_u + (size_t)(mbase + rhi) * NCP + colg)
                 : (Go   + (size_t)(mbase + rhi) * NCP + (colg - 64));
#pragma unroll
    for (int r = 0; r < 8; ++r) {
        g[(size_t)(r)      * NCP] = acc0[r];
        g[(size_t)(r + 16) * NCP] = acc1[r];
        g[(size_t)(r + 32) * NCP] = acc2[r];
        g[(size_t)(r + 48) * NCP] = acc3[r];
    }
}

// ---------- kernel 4: pseudo-labels for unlabeled rows ----------
__global__ __launch_bounds__(256) void k_pseudo(const float* __restrict__ Go,
                                                const float* __restrict__ b_o,
                                                int* __restrict__ pred_u,
                                                float* __restrict__ wmid,
                                                float* __restrict__ wtail) {
    int u = blockIdx.x * 256 + threadIdx.x;
    if (u >= N_U) return;
    const float* g = Go + (size_t)u * NCP;
    float m = -3.4e38f; int am = 0;
    for (int c = 0; c < NC; ++c) {
        float z = g[c] + b_o[c];
        if (z > m) { m = z; am = c; }
    }
    float s = 0.0f;
    for (int c = 0; c < NC; ++c) {
        float z = g[c] + b_o[c];
        s += expf(z - m);
    }
    float score = 1.0f / s;                       // max softmax prob
    pred_u[u] = am;
    wmid[u]  = (am >= 16 && am < 36 && score > 0.5f) ? 1.0f : 0.0f;
    wtail[u] = (am >= 36 && score > 0.3f) ? 1.0f : 0.0f;
}

// ---------- kernel 5: weighted CE over 5*N_U augmented samples ----------
__global__ __launch_bounds__(256) void k_ce(const float* __restrict__ Gw,
                                            const float* __restrict__ bvec,
                                            const int* __restrict__ label,
                                            const int* __restrict__ idx_m,
                                            const int* __restrict__ idx_t,
                                            const int* __restrict__ pred_u,
                                            const float* __restrict__ wmid,
                                            const float* __restrict__ wtail,
                                            float* __restrict__ partial) {
    __shared__ float s_t[256];
    __shared__ float s_c[256];
    int s = blockIdx.x * 256 + threadIdx.x;       // 0 .. 5*N_U-1
    int u = s & (N_U - 1);
    int j = s >> 14;                              // 0..4
    float w; int idx;
    if (j < 2) { w = wmid[u];  idx = idx_m[j * N_U + u]; }
    else       { w = wtail[u]; idx = idx_t[(j - 2) * N_U + u]; }

    float ce = 0.0f;
    if (w > 0.0f) {
        const float* go = Gw + (size_t)idx * NCP;
        const float* gu = Gw + (size_t)(N_O + u) * NCP;
        int lab = label[idx];
        int pp  = pred_u[u];
        float m = -3.4e38f;
        for (int c = 0; c < NC; ++c) {
            float z = 0.7f * go[c] + 0.3f * gu[c] + bvec[c];
            m = fmaxf(m, z);
        }
        float sum = 0.0f, zl = 0.0f, zp = 0.0f;
        for (int c = 0; c < NC; ++c) {
            float z = 0.7f * go[c] + 0.3f * gu[c] + bvec[c];
            sum += expf(z - m);
            if (c == lab) zl = z;
            if (c == pp)  zp = z;
        }
        ce = (m + logf(sum)) - 0.7f * zl - 0.3f * zp;
    }
    s_t[threadIdx.x] = ce;
    s_c[threadIdx.x] = w;
    __syncthreads();
    for (int off = 128; off > 0; off >>= 1) {
        if (threadIdx.x < off) {
            s_t[threadIdx.x] += s_t[threadIdx.x + off];
            s_c[threadIdx.x] += s_c[threadIdx.x + off];
        }
        __syncthreads();
    }
    if (threadIdx.x == 0) {
        partial[2 * blockIdx.x]     = s_t[0];
        partial[2 * blockIdx.x + 1] = s_c[0];
    }
}

// ---------- kernel 6: final reduction ----------
__global__ __launch_bounds__(256) void k_final(const float* __restrict__ partial,
                                               int nb, float* __restrict__ out) {
    __shared__ float s_t[256];
    __shared__ float s_c[256];
    float t = 0.0f, c = 0.0f;
    for (int i = threadIdx.x; i < nb; i += 256) {
        t += partial[2 * i];
        c += partial[2 * i + 1];
    }
    s_t[threadIdx.x] = t; s_c[threadIdx.x] = c;
    __syncthreads();
    for (int off = 128; off > 0; off >>= 1) {
        if (threadIdx.x < off) {
            s_t[threadIdx.x] += s_t[threadIdx.x + off];
            s_c[threadIdx.x] += s_c[threadIdx.x + off];
        }
        __syncthreads();
    }
    if (threadIdx.x == 0) out[0] = s_t[0] / fmaxf(s_c[0], 1.0f);
}

extern "C" void kernel_launch(void* const* d_in, const int* in_sizes, int n_in,
                              void* d_out, int out_size, void* d_ws, size_t ws_size,
                              hipStream_t stream) {
    const float* feat  = (const float*)d_in[0];
    const int*   label = (const int*)d_in[1];
    const float* W_o   = (const float*)d_in[2];
    const float* b_o   = (const float*)d_in[3];
    const float* W     = (const float*)d_in[4];
    const float* bvec  = (const float*)d_in[5];
    // d_in[6], d_in[7] (group masks) are deterministic functions of class idx; computed analytically.
    const int*   idx_m = (const int*)d_in[8];
    const int*   idx_t = (const int*)d_in[9];
    float* out = (float*)d_out;

    char* ws = (char*)d_ws;
    float* Gw      = (float*)(ws + 0);          // 32768*64 f32 = 8388608 B
    float* Go      = (float*)(ws + 8388608);    // 16384*64 f32 = 4194304 B
    float* Wcat    = (float*)(ws + 12582912);   // 128*1024 f32 = 524288 B
    int*   pred_u  = (int*)  (ws + 13107200);   // 65536 B
    float* wmid    = (float*)(ws + 13172736);   // 65536 B
    float* wtail   = (float*)(ws + 13238272);   // 65536 B
    float* partial = (float*)(ws + 13303808);   // 320*2 f32

    // 1) build concatenated padded weights (128*1024 elements)
    k_pad_weights<<<512, 256, 0, stream>>>(W, W_o, Wcat);
    // 2) o-rows: Gw[0:16384] = feat_o @ W^T  (4 waves/block, 64 rows/block)
    k_gemm_wmma<<<N_O / 64, 128, 0, stream>>>(feat, Wcat, Gw);
    // 3) u-rows: single pass over feat_u computes both heads (8 waves/block)
    k_gemm_wmma_dual<<<N_U / 64, 256, 0, stream>>>(feat + (size_t)N_O * DD, Wcat,
                                                   Gw + (size_t)N_O * NCP, Go);
    // 4) pseudo labels
    k_pseudo<<<N_U / 256, 256, 0, stream>>>(Go, b_o, pred_u, wmid, wtail);
    // 5) CE over 5*N_U samples -> 320 partial pairs
    k_ce<<<(5 * N_U) / 256, 256, 0, stream>>>(Gw, bvec, label, idx_m, idx_t,
                                              pred_u, wmid, wtail, partial);
    // 6) final scalar
    k_final<<<1, 256, 0, stream>>>(partial, (5 * N_U) / 256, out);
}